// customlayer_28475633172946
// MI455X (gfx1250) — compile-verified
//
#include <hip/hip_runtime.h>

typedef float v2f __attribute__((ext_vector_type(2)));
typedef float v8f __attribute__((ext_vector_type(8)));

#define IMG_H 256
#define IMG_W 512
#define OUT_H 32
#define OUT_W 256
#define KT_ROWS 287          // di = i-ii+255 in [0,286]
#define KT_STRIDE 768        // 767 real cols + 1 zero pad col (contraction padded to 768)
#define P_GUARD 8            // zero guard in front of padded image row (B reads t-1 down to -1)
#define P_ELEMS (P_GUARD + 1024)          // padded row: zeros[0,255) data[255,767) zeros[767,1024)
#define KROWS_ELEMS (32 * KT_STRIDE)      // 32 kernel rows staged in LDS per workgroup
#define SMEM_ELEMS (KROWS_ELEMS + P_ELEMS)

// ---------------- Kernel 1: build shift-invariant weight table --------------
// Ktab[a][s] = 1/((a-255+pos0)^2 + (s-511+pos1)^2), col 767 zeroed for K-padding.
__global__ void build_ktab(const float* __restrict__ pos, float* __restrict__ ktab) {
    int idx = blockIdx.x * blockDim.x + threadIdx.x;
    if (idx >= KT_ROWS * KT_STRIDE) return;
    int a = idx / KT_STRIDE;
    int s = idx - a * KT_STRIDE;
    float v = 0.0f;
    if (s < 767) {
        float dx = (float)(a - 255) + pos[0];
        float dy = (float)(s - 511) + pos[1];
        v = 1.0f / (dx * dx + dy * dy);
    }
    ktab[idx] = v;
}

// ---------------- Kernel 2: split-K GEMM via f32 WMMA -----------------------
// One block per image row ii. Output D[i,j] partial = sum_s Krows[i][s]*P[j-s+766].
// WMMA f32 16x16x4 layouts (wave32):
//   A(16x4): lane -> M = lane%16, vgpr v -> K = v + 2*(lane/16)   (float2/lane)
//   B(4x16): lane -> N = lane%16, vgpr v -> K = v + 2*(lane/16)
//   C/D(16x16): lane -> N = lane%16, vgpr r -> M = r + 8*(lane/16)
__global__ void __launch_bounds__(256)
conv_wmma(const float* __restrict__ img,
          const float* __restrict__ ktab,
          float* __restrict__ part) {
    extern __shared__ float smem[];
    float* Krows = smem;                                  // [32][768]
    float* Pp    = smem + KROWS_ELEMS + P_GUARD;          // logical t index, valid [-8, 1024)

    const int tid = threadIdx.x;
    const int ii  = blockIdx.x;

    // Stage zero-padded image row: Pp[t] = img[ii][t-255] for t in [255,767), else 0.
    for (int p = tid; p < P_ELEMS; p += 256) {
        int x = p - (P_GUARD + 255);
        float v = (x >= 0 && x < IMG_W) ? img[ii * IMG_W + x] : 0.0f;
        smem[KROWS_ELEMS + p] = v;
    }
    // Stage the 32 kernel rows this ii needs: row(i) = i - ii + 255 (always in [0,287)).
    for (int i = 0; i < 32; ++i) {
        const float* g = ktab + (i - ii + 255) * KT_STRIDE;
        float* l = Krows + i * KT_STRIDE;
        for (int s = tid; s < KT_STRIDE; s += 256) l[s] = g[s];
    }
    __syncthreads();

    const int lane = tid & 31;
    const int wv   = tid >> 5;        // wave 0..7 -> N-tiles {2wv, 2wv+1}
    const int l16  = lane & 15;
    const int h    = lane >> 4;       // 0/1 selects K pair {0,1} vs {2,3}

    const float* krow0 = Krows + l16 * KT_STRIDE + 2 * h;          // M-tile 0 (i = 0..15)
    const float* krow1 = Krows + (16 + l16) * KT_STRIDE + 2 * h;   // M-tile 1 (i = 16..31)
    const int jA  = (2 * wv) * 16 + l16;
    const int jB  = jA + 16;
    const int tA0 = jA - 2 * h + 766;   // B element for k: Pp[t0 - v], v = k - 2h
    const int tB0 = jB - 2 * h + 766;

    v8f acc00 = {}; v8f acc01 = {}; v8f acc10 = {}; v8f acc11 = {};

    for (int s0 = 0; s0 < KT_STRIDE; s0 += 4) {
        v2f a0 = *(const v2f*)(krow0 + s0);
        v2f a1 = *(const v2f*)(krow1 + s0);
        int tA = tA0 - s0, tB = tB0 - s0;
        v2f b0, b1;
        b0.x = Pp[tA]; b0.y = Pp[tA - 1];
        b1.x = Pp[tB]; b1.y = Pp[tB - 1];
        acc00 = __builtin_amdgcn_wmma_f32_16x16x4_f32(false, a0, false, b0, (short)0, acc00, false, false);
        acc01 = __builtin_amdgcn_wmma_f32_16x16x4_f32(false, a0, false, b1, (short)0, acc01, false, false);
        acc10 = __builtin_amdgcn_wmma_f32_16x16x4_f32(false, a1, false, b0, (short)0, acc10, false, false);
        acc11 = __builtin_amdgcn_wmma_f32_16x16x4_f32(false, a1, false, b1, (short)0, acc11, false, false);
    }

    // Write this ii's 32x256 partial (every element covered exactly once; no atomics).
    float* base = part + (size_t)ii * (OUT_H * OUT_W);
    #pragma unroll
    for (int r = 0; r < 8; ++r) {
        int i0 = 8 * h + r;
        int i1 = 16 + 8 * h + r;
        base[i0 * OUT_W + jA] = acc00[r];
        base[i0 * OUT_W + jB] = acc01[r];
        base[i1 * OUT_W + jA] = acc10[r];
        base[i1 * OUT_W + jB] = acc11[r];
    }
}

// ---------------- Kernel 3: deterministic split-K reduction -----------------
__global__ void reduce_parts(const float* __restrict__ part, float* __restrict__ out) {
    int o = blockIdx.x * blockDim.x + threadIdx.x;
    if (o >= OUT_H * OUT_W) return;
    float s = 0.0f;
    for (int w = 0; w < IMG_H; ++w) s += part[(size_t)w * (OUT_H * OUT_W) + o];
    out[o] = s;
}

extern "C" void kernel_launch(void* const* d_in, const int* in_sizes, int n_in,
                              void* d_out, int out_size, void* d_ws, size_t ws_size,
                              hipStream_t stream) {
    const float* img = (const float*)d_in[0];   // [256,512] f32
    const float* pos = (const float*)d_in[1];   // [2] f32
    float* out  = (float*)d_out;                // [32,256] f32

    float* ktab = (float*)d_ws;                 // 287*768 f32 (~880 KB)
    float* part = (float*)d_ws + (1 << 18);     // 1 MB offset; 256 * 8192 f32 (8 MB)

    build_ktab<<<(KT_ROWS * KT_STRIDE + 255) / 256, 256, 0, stream>>>(pos, ktab);

    size_t smem_bytes = SMEM_ELEMS * sizeof(float);   // 102,432 B of the WGP's 320 KB
    conv_wmma<<<IMG_H, 256, smem_bytes, stream>>>(img, ktab, part);

    reduce_parts<<<(OUT_H * OUT_W + 255) / 256, 256, 0, stream>>>(part, out);
}